// DSTGNN_85315230368166
// MI455X (gfx1250) — compile-verified
//
#include <hip/hip_runtime.h>
#include <hip/hip_bf16.h>

typedef __attribute__((ext_vector_type(2))) float v2f;
typedef __attribute__((ext_vector_type(8))) float v8f;

#define MROWS (2 * 12 * 512)          // 12288 rows for (b,t,n)
#define BN    (2 * 512)               // 1024 rows for (b,n)

// ---------------------------------------------------------------------------
// Generic dual-input GEMM, 64 output cols:
//   out[M x 64] = act( A1[M x K1] @ W1[K1 x 64] + A2[M x K2] @ W2[K2 x 64] + bias )
// One wave computes one 16x16 tile via V_WMMA_F32_16X16X4_F32.
// Block = 256 threads = 8 waves = 32 rows x 64 cols. grid.x = M/32.
// ---------------------------------------------------------------------------
__global__ void gemm_dual64(const float* __restrict__ A1, int K1, const float* __restrict__ W1,
                            const float* __restrict__ A2, int K2, const float* __restrict__ W2,
                            const float* __restrict__ bias, float* __restrict__ out,
                            int relu)
{
    const int lane  = threadIdx.x & 31;
    const int wave  = threadIdx.x >> 5;
    const int mTile = blockIdx.x * 32 + (wave >> 2) * 16;
    const int nTile = (wave & 3) * 16;
    const int m  = mTile + (lane & 15);     // A row for this lane
    const int n  = nTile + (lane & 15);     // B col for this lane
    const int kh = (lane >> 4) * 2;         // k sub-offset within 4-step

    v8f acc = {0.f, 0.f, 0.f, 0.f, 0.f, 0.f, 0.f, 0.f};

    const float* a1 = A1 + (long)m * K1 + kh;
    const float* w1 = W1 + kh * 64 + n;
    for (int k0 = 0; k0 < K1; k0 += 4) {
        v2f a = { a1[k0], a1[k0 + 1] };
        v2f b = { w1[k0 * 64], w1[k0 * 64 + 64] };
        acc = __builtin_amdgcn_wmma_f32_16x16x4_f32(false, a, false, b, (short)0, acc, false, false);
    }
    if (K2 > 0) {
        const float* a2 = A2 + (long)m * K2 + kh;
        const float* w2 = W2 + kh * 64 + n;
        for (int k0 = 0; k0 < K2; k0 += 4) {
            v2f a = { a2[k0], a2[k0 + 1] };
            v2f b = { w2[k0 * 64], w2[k0 * 64 + 64] };
            acc = __builtin_amdgcn_wmma_f32_16x16x4_f32(false, a, false, b, (short)0, acc, false, false);
        }
    }

    const float bv = bias[n];
    const int rbase = mTile + (lane >> 4) * 8;
    #pragma unroll
    for (int i = 0; i < 8; ++i) {
        float v = acc[i] + bv;
        if (relu) v = fmaxf(v, 0.f);
        out[(long)(rbase + i) * 64 + n] = v;
    }
}

// ---------------------------------------------------------------------------
// TCN weight pre-transpose: W[i][o][c][t] (12,128,64,12) -> Wt[i][t*64+c][o],
// applying causal mask (zero where t > i).
// ---------------------------------------------------------------------------
__global__ void tcn_wt(const float* __restrict__ W, float* __restrict__ Wt, int causal)
{
    int idx = blockIdx.x * 256 + threadIdx.x;     // 12*768*128 = 1179648 total
    if (idx >= 12 * 768 * 128) return;
    int o = idx & 127;
    int k = (idx >> 7) % 768;
    int i = idx / (768 * 128);
    int t = k >> 6, c = k & 63;
    float w = W[(((long)i * 128 + o) * 64 + c) * 12 + t];
    if (causal && t > i) w = 0.f;
    Wt[idx] = w;
}

// ---------------------------------------------------------------------------
// TCN GEMM: raw[b,i,n,o] = sum_{t,c} h[b,t,n,c] * Wt[i][(t,c)][o] + bias[i,o]
// rows r = b*512+n (1024), cols o (128).  grid = (64, 12), block 256 (8 waves,
// one 16-col tile each).  K = 64*(i+1) if causal else 768.
// ---------------------------------------------------------------------------
__global__ void tcn_gemm(const float* __restrict__ h, const float* __restrict__ Wt,
                         const float* __restrict__ bias, float* __restrict__ raw, int causal)
{
    const int i = blockIdx.y;
    const int K = causal ? 64 * (i + 1) : 768;
    const int lane  = threadIdx.x & 31;
    const int wave  = threadIdx.x >> 5;
    const int mTile = blockIdx.x * 16;
    const int nTile = wave * 16;
    const int r = mTile + (lane & 15);
    const int b = r >> 9, n = r & 511;
    const int kh  = (lane >> 4) * 2;
    const int col = nTile + (lane & 15);

    const float* ab = h + (long)b * (12 * 512 * 64) + n * 64;
    const float* wb = Wt + (long)i * (768 * 128) + kh * 128 + col;

    v8f acc = {0.f, 0.f, 0.f, 0.f, 0.f, 0.f, 0.f, 0.f};
    for (int k0 = 0; k0 < K; k0 += 4) {
        int k = k0 + kh;                       // even, so k,k+1 share same t
        int off = (k >> 6) * (512 * 64) + (k & 63);
        v2f a = { ab[off], ab[off + 1] };
        v2f bvec = { wb[k0 * 128], wb[k0 * 128 + 128] };
        acc = __builtin_amdgcn_wmma_f32_16x16x4_f32(false, a, false, bvec, (short)0, acc, false, false);
    }

    const float bv = bias[i * 128 + col];
    const int rbase = mTile + (lane >> 4) * 8;
    #pragma unroll
    for (int j = 0; j < 8; ++j) {
        int rr = rbase + j;
        int bb = rr >> 9, nn = rr & 511;
        raw[(((long)bb * 12 + i) * 512 + nn) * 128 + col] = acc[j] + bv;
    }
}

// GLU: out[...,c] = raw[...,c] * sigmoid(raw[...,64+c])
__global__ void glu(const float* __restrict__ raw, float* __restrict__ out)
{
    int idx = blockIdx.x * 256 + threadIdx.x;    // 786432 total
    int c = idx & 63, rn = idx >> 6;
    float l = raw[(long)rn * 128 + c];
    float r = raw[(long)rn * 128 + 64 + c];
    out[idx] = l * (1.f / (1.f + __expf(-r)));
}

// ---------------------------------------------------------------------------
// MHA: relu(q k^T), + I, row-normalize, times v.  One block handles half the
// rows of one (b,t,head): grid.x = 24*8*2 = 384, block 256.  k/v head tiles
// (512x8 each) staged in LDS; all lanes stream the same m -> LDS broadcast.
// ---------------------------------------------------------------------------
__global__ void mha(const float* __restrict__ q, const float* __restrict__ k,
                    const float* __restrict__ v, float* __restrict__ out)
{
    __shared__ float ks[512 * 8];
    __shared__ float vs[512 * 8];
    const int blk  = blockIdx.x;
    const int half = blk & 1;
    const int hd   = (blk >> 1) & 7;
    const int bt   = blk >> 4;                  // (b*12 + t) in [0,24)
    const float* kb = k + (long)bt * 512 * 64 + hd * 8;
    const float* vb = v + (long)bt * 512 * 64 + hd * 8;

    for (int idx = threadIdx.x; idx < 512 * 8; idx += 256) {
        int m = idx >> 3, d = idx & 7;
        ks[idx] = kb[(long)m * 64 + d];
        vs[idx] = vb[(long)m * 64 + d];
    }
    __syncthreads();

    const int n = half * 256 + threadIdx.x;
    const float* qn = q + (long)bt * 512 * 64 + (long)n * 64 + hd * 8;
    float qr[8];
    #pragma unroll
    for (int d = 0; d < 8; ++d) qr[d] = qn[d];

    float o[8] = {0.f, 0.f, 0.f, 0.f, 0.f, 0.f, 0.f, 0.f};
    float ssum = 0.f;
    for (int m = 0; m < 512; ++m) {
        float s = 0.f;
        #pragma unroll
        for (int d = 0; d < 8; ++d) s += qr[d] * ks[m * 8 + d];
        s = fmaxf(s, 0.f);
        if (m == n) s += 1.f;
        ssum += s;
        #pragma unroll
        for (int d = 0; d < 8; ++d) o[d] += s * vs[m * 8 + d];
    }
    const float inv = 1.f / ssum;
    float* op = out + (long)bt * 512 * 64 + (long)n * 64 + hd * 8;
    #pragma unroll
    for (int d = 0; d < 8; ++d) op[d] = o[d] * inv;
}

// ---------------------------------------------------------------------------
// LayerNorm(x + res) * g + b over C=64.  One wave per row (2 elems/lane),
// wave32 shuffle reductions.  grid = 12288/8 blocks of 256.
// ---------------------------------------------------------------------------
__global__ void ln_res(const float* __restrict__ x, const float* __restrict__ res,
                       const float* __restrict__ g, const float* __restrict__ be,
                       float* __restrict__ out)
{
    const int lane = threadIdx.x & 31;
    const int row  = (blockIdx.x * blockDim.x + threadIdx.x) >> 5;
    const float* xr = x + (long)row * 64;
    const float* rr = res + (long)row * 64;
    float v0 = xr[lane] + rr[lane];
    float v1 = xr[lane + 32] + rr[lane + 32];
    float s = v0 + v1;
    for (int off = 16; off > 0; off >>= 1) s += __shfl_xor(s, off, 32);
    const float mu = s * (1.f / 64.f);
    float d0 = v0 - mu, d1 = v1 - mu;
    float vv = d0 * d0 + d1 * d1;
    for (int off = 16; off > 0; off >>= 1) vv += __shfl_xor(vv, off, 32);
    const float inv = rsqrtf(vv * (1.f / 64.f) + 1e-5f);
    out[(long)row * 64 + lane]      = d0 * inv * g[lane] + be[lane];
    out[(long)row * 64 + lane + 32] = d1 * inv * g[lane + 32] + be[lane + 32];
}

// start layer 0: h0[r,c] = relu(x[r] * W1[c] + b1[c])
__global__ void start1(const float* __restrict__ x, const float* __restrict__ W1,
                       const float* __restrict__ b1, float* __restrict__ h0)
{
    int idx = blockIdx.x * 256 + threadIdx.x;    // 786432
    int c = idx & 63, r = idx >> 6;
    h0[idx] = fmaxf(x[r] * W1[c] + b1[c], 0.f);
}

// time-embedding FF: one-hot(dow) / one-hot(tod) -> row picks of W1, then C->C
__global__ void te_ff(const int* __restrict__ te, const float* __restrict__ W1,
                      const float* __restrict__ b1, const float* __restrict__ W2,
                      const float* __restrict__ b2, float* __restrict__ te_out)
{
    __shared__ float tmp[64];
    const int bt = blockIdx.x;                   // (b*24 + t) in [0,48)
    const int c  = threadIdx.x;
    const int dow = ((unsigned)te[bt * 2 + 0]) % 7u;
    const int tod = ((unsigned)te[bt * 2 + 1]) % 288u;
    float v = W1[dow * 64 + c] + W1[(7 + tod) * 64 + c] + b1[c];
    tmp[c] = fmaxf(v, 0.f);
    __syncthreads();
    float s = b2[c];
    for (int j = 0; j < 64; ++j) s += tmp[j] * W2[j * 64 + c];
    te_out[bt * 64 + c] = s;
}

// step/stef: ste[b,t,n,c] = se_out[n,c] + te_out[b,(t | 12+t),c]
__global__ void ste_build(const float* __restrict__ se_o, const float* __restrict__ te_o,
                          float* __restrict__ step, float* __restrict__ stef)
{
    int idx = blockIdx.x * 256 + threadIdx.x;    // 786432
    int c = idx & 63;
    int n = (idx >> 6) & 511;
    int bt = idx >> 15;                          // b*12 + t
    int t = bt % 12, b = bt / 12;
    float se = se_o[n * 64 + c];
    step[idx] = se + te_o[(b * 24 + t) * 64 + c];
    stef[idx] = se + te_o[(b * 24 + 12 + t) * 64 + c];
}

// end layer 1: out[r] = dot(e1[r,:], W2[:,0]) + b2[0]
__global__ void end2(const float* __restrict__ e1, const float* __restrict__ W2,
                     const float* __restrict__ b2, float* __restrict__ out)
{
    const int lane = threadIdx.x & 31;
    const int row  = (blockIdx.x * blockDim.x + threadIdx.x) >> 5;
    float s = e1[(long)row * 64 + lane] * W2[lane]
            + e1[(long)row * 64 + 32 + lane] * W2[32 + lane];
    for (int off = 16; off > 0; off >>= 1) s += __shfl_xor(s, off, 32);
    if (lane == 0) out[row] = s + b2[0];
}

// ---------------------------------------------------------------------------
// Host-side orchestration
// ---------------------------------------------------------------------------
static void dense(const float* A1, int K1, const float* W1,
                  const float* A2, int K2, const float* W2,
                  const float* bias, float* out, int M, int relu, hipStream_t s)
{
    gemm_dual64<<<M / 32, 256, 0, s>>>(A1, K1, W1, A2, K2, W2, bias, out, relu);
}

// P: 18 param arrays (kt W,b | vt W,b | qfc W,b | kfc W,b | vfc W,b | ln g,b | ff W1,b1,W2,b2 | ffln g,b)
static void run_sagcn(const float* const* P, float* h, const float* ste,
                      float* Wt, float* raw, float* kx, float* vx,
                      float* q, float* kk, float* vv, float* t1, float* t2,
                      hipStream_t s)
{
    tcn_wt<<<4608, 256, 0, s>>>(P[0], Wt, 1);
    tcn_gemm<<<dim3(64, 12), 256, 0, s>>>(h, Wt, P[1], raw, 1);
    glu<<<3072, 256, 0, s>>>(raw, kx);
    tcn_wt<<<4608, 256, 0, s>>>(P[2], Wt, 1);
    tcn_gemm<<<dim3(64, 12), 256, 0, s>>>(h, Wt, P[3], raw, 1);
    glu<<<3072, 256, 0, s>>>(raw, vx);
    dense(h,  64, P[4], ste, 64, P[4] + 4096, P[5], q,  MROWS, 0, s);
    dense(kx, 64, P[6], ste, 64, P[6] + 4096, P[7], kk, MROWS, 0, s);
    dense(vx, 64, P[8], ste, 64, P[8] + 4096, P[9], vv, MROWS, 0, s);
    mha<<<384, 256, 0, s>>>(q, kk, vv, t1);
    ln_res<<<1536, 256, 0, s>>>(t1, h, P[10], P[11], t2);
    dense(t2, 64, P[12], nullptr, 0, nullptr, P[13], t1, MROWS, 1, s);
    dense(t1, 64, P[14], nullptr, 0, nullptr, P[15], q,  MROWS, 0, s);
    ln_res<<<1536, 256, 0, s>>>(q, t2, P[16], P[17], h);
}

// P: 16 param arrays (vt W,b | qfc W,b | kfc W,b | vfc W,b | ln g,b | ff W1,b1,W2,b2 | ffln g,b)
static void run_eda(const float* const* P, float* h, const float* step, const float* stef,
                    float* Wt, float* raw, float* vx,
                    float* q, float* kk, float* vv, float* t1, float* t2,
                    hipStream_t s)
{
    tcn_wt<<<4608, 256, 0, s>>>(P[0], Wt, 0);
    tcn_gemm<<<dim3(64, 12), 256, 0, s>>>(h, Wt, P[1], raw, 0);
    glu<<<3072, 256, 0, s>>>(raw, vx);
    dense(stef, 64, P[2], nullptr, 0, nullptr, P[3], q,  MROWS, 0, s);
    dense(step, 64, P[4], nullptr, 0, nullptr, P[5], kk, MROWS, 0, s);
    dense(vx,   64, P[6], nullptr, 0, nullptr, P[7], vv, MROWS, 0, s);
    mha<<<384, 256, 0, s>>>(q, kk, vv, t1);
    ln_res<<<1536, 256, 0, s>>>(t1, h, P[8], P[9], t2);
    dense(t2, 64, P[10], nullptr, 0, nullptr, P[11], t1, MROWS, 1, s);
    dense(t1, 64, P[12], nullptr, 0, nullptr, P[13], q,  MROWS, 0, s);
    ln_res<<<1536, 256, 0, s>>>(q, t2, P[14], P[15], h);
}

extern "C" void kernel_launch(void* const* d_in, const int* in_sizes, int n_in,
                              void* d_out, int out_size, void* d_ws, size_t ws_size,
                              hipStream_t stream)
{
    (void)in_sizes; (void)n_in; (void)out_size; (void)ws_size;
    auto F = [&](int i) { return (const float*)d_in[i]; };
    auto PP = [&](int base) { return reinterpret_cast<const float* const*>(d_in + base); };

    const float* x  = (const float*)d_in[0];   // (2,12,512)
    const int*   te = (const int*)d_in[1];     // (2,24,2)
    // params: 2:se  3-6:ff_se  7-10:ff_te  11-14:start  15-18:end
    //         19..36:enc0  37..54:enc1  55..72:dec0  73..90:dec1  91..106:eda

    float* w = (float*)d_ws;
    float* se_tmp = w; w += 512 * 64;
    float* se_out = w; w += 512 * 64;
    float* te_out = w; w += 2 * 24 * 64;
    float* step   = w; w += MROWS * 64;
    float* stef   = w; w += MROWS * 64;
    float* h      = w; w += MROWS * 64;
    float* Wt     = w; w += 12 * 768 * 128;
    float* raw    = w; w += MROWS * 128;
    float* kx     = w; w += MROWS * 64;
    float* vx     = w; w += MROWS * 64;
    float* q      = w; w += MROWS * 64;
    float* kk     = w; w += MROWS * 64;
    float* vv     = w; w += MROWS * 64;
    float* t1     = w; w += MROWS * 64;
    float* t2     = w; w += MROWS * 64;

    // spatial + temporal embeddings
    dense(F(2), 64, F(3), nullptr, 0, nullptr, F(4), se_tmp, 512, 1, stream);
    dense(se_tmp, 64, F(5), nullptr, 0, nullptr, F(6), se_out, 512, 0, stream);
    te_ff<<<48, 64, 0, stream>>>(te, F(7), F(8), F(9), F(10), te_out);
    ste_build<<<3072, 256, 0, stream>>>(se_out, te_out, step, stef);

    // start FF: (B,T,N,1) -> (B,T,N,64)
    start1<<<3072, 256, 0, stream>>>(x, F(11), F(12), t1);
    dense(t1, 64, F(13), nullptr, 0, nullptr, F(14), h, MROWS, 0, stream);

    // encoder / bridge / decoder
    run_sagcn(PP(19), h, step, Wt, raw, kx, vx, q, kk, vv, t1, t2, stream);
    run_sagcn(PP(37), h, step, Wt, raw, kx, vx, q, kk, vv, t1, t2, stream);
    run_eda  (PP(91), h, step, stef, Wt, raw, vx, q, kk, vv, t1, t2, stream);
    run_sagcn(PP(55), h, stef, Wt, raw, kx, vx, q, kk, vv, t1, t2, stream);
    run_sagcn(PP(73), h, stef, Wt, raw, kx, vx, q, kk, vv, t1, t2, stream);

    // end FF: (B,T,N,64) -> (B,T,N)
    dense(h, 64, F(15), nullptr, 0, nullptr, F(16), t1, MROWS, 1, stream);
    end2<<<1536, 256, 0, stream>>>(t1, F(17), F(18), (float*)d_out);
}